// Gamma_Net_79164837199902
// MI455X (gfx1250) — compile-verified
//
#include <hip/hip_runtime.h>
#include <math.h>

#define N_NODES 80000
#define N_EDGES 1280000
#define N_GRAPHS 256
#define E_TOT (N_EDGES + N_NODES)
#define NEG_SLOPE 0.2f

typedef float v2f __attribute__((ext_vector_type(2)));
typedef float v8f __attribute__((ext_vector_type(8)));

// ---------------------------------------------------------------------------
// FP32 GEMM on the CDNA5 matrix pipe: one wave per 16x16 C tile, K stepped by 4
// using V_WMMA_F32_16X16X4_F32.  K and N are compile-time so the main K-loop
// has no bounds checks (branch-free, unconditional b32 loads) and unrolls.
//   A lane layout (16x4):  lane = 16*hi + m  holds A[m][2*hi + v], v = 0,1
//   B lane layout (4x16):  lane = 16*hi + n  holds B[2*hi + v][n], v = 0,1
//   D lane layout (16x16): lane = 16*hi + n  holds D[v + 8*hi][n], v = 0..7
// M must be a multiple of 16 (N_NODES = 80000 = 5000*16).
// ---------------------------------------------------------------------------
template <int K, int N>
__global__ void wmma_gemm_f32(const float* __restrict__ A, const float* __restrict__ B,
                              float* __restrict__ C, int M) {
  constexpr int TILES_N = (N + 15) / 16;
  const int lane = threadIdx.x & 31;
  const int wave = (blockIdx.x * blockDim.x + threadIdx.x) >> 5;
  const int tiles_m = M >> 4;
  if (wave >= tiles_m * TILES_N) return;
  const int tm = (wave / TILES_N) << 4;
  const int tn = (wave % TILES_N) << 4;
  const int lrow = lane & 15;   // m for A, n for B/D
  const int lhi  = lane >> 4;   // half-wave select
  const int ar = tm + lrow;                 // always < M
  const int bn = tn + lrow;
  const bool inb = (bn < N);
  const int bnc = inb ? bn : 0;             // clamped -> loads always legal
  const float* __restrict__ Arow = A + (size_t)ar * K;
  v8f acc = {};

  constexpr int KM = K & ~3;                // full 4-wide K blocks
#pragma unroll 4
  for (int k0 = 0; k0 < KM; k0 += 4) {
    const int ka = k0 + (lhi << 1);
    v2f a, b;
    a.x = Arow[ka];
    a.y = Arow[ka + 1];
    const float b0 = B[ka * N + bnc];
    const float b1 = B[(ka + 1) * N + bnc];
    b.x = inb ? b0 : 0.f;
    b.y = inb ? b1 : 0.f;
    acc = __builtin_amdgcn_wmma_f32_16x16x4_f32(false, a, false, b,
                                                (short)0, acc, false, false);
  }
  if constexpr ((K & 3) != 0) {             // K-remainder: clamp + select, no branch
    const int ka = KM + (lhi << 1);
    const bool k0ok = (ka < K);
    const bool k1ok = (ka + 1 < K);
    const int ka0 = k0ok ? ka : 0;
    const int ka1 = k1ok ? ka + 1 : 0;
    v2f a, b;
    const float a0 = Arow[ka0];
    const float a1 = Arow[ka1];
    const float b0 = B[ka0 * N + bnc];
    const float b1 = B[ka1 * N + bnc];
    a.x = k0ok ? a0 : 0.f;
    a.y = k1ok ? a1 : 0.f;
    b.x = (inb && k0ok) ? b0 : 0.f;
    b.y = (inb && k1ok) ? b1 : 0.f;
    acc = __builtin_amdgcn_wmma_f32_16x16x4_f32(false, a, false, b,
                                                (short)0, acc, false, false);
  }
#pragma unroll
  for (int v = 0; v < 8; ++v) {
    const int r = tm + v + (lhi << 3);      // always < M
    const int c = tn + lrow;
    if (c < N) C[(size_t)r * N + c] = acc[v];
  }
}

// al_src[n,h] = sum_c h[n,h,c]*a_src[h,c]; same for al_dst
__global__ void node_logits(const float* __restrict__ h, const float* __restrict__ a_src,
                            const float* __restrict__ a_dst, float* __restrict__ al_s,
                            float* __restrict__ al_d, int n, int heads, int C) {
  int idx = blockIdx.x * blockDim.x + threadIdx.x;
  if (idx >= n * heads) return;
  int node = idx / heads, hd = idx - node * heads;
  const float* hp = h + ((size_t)node * heads + hd) * C;
  const float* as = a_src + hd * C;
  const float* ad = a_dst + hd * C;
  float ss = 0.f, sd = 0.f;
  for (int c = 0; c < C; ++c) {
    float v = hp[c];
    ss = fmaf(v, as[c], ss);
    sd = fmaf(v, ad[c], sd);
  }
  al_s[idx] = ss;
  al_d[idx] = sd;
}

__global__ void fill_f32(float* __restrict__ p, float v, int n) {
  int i = blockIdx.x * blockDim.x + threadIdx.x;
  if (i < n) p[i] = v;
}

// alpha = leaky_relu(al_src[src]+al_dst[dst]); segment max over dst via f32 atomic max
__global__ void edge_alpha_max(const long long* __restrict__ esrc,
                               const long long* __restrict__ edst,
                               const float* __restrict__ al_s, const float* __restrict__ al_d,
                               float* __restrict__ alpha, float* __restrict__ mx, int heads) {
  int e = blockIdx.x * blockDim.x + threadIdx.x;
  if (e >= E_TOT) return;
  int s, d;
  if (e < N_EDGES) { s = (int)esrc[e]; d = (int)edst[e]; }
  else             { s = d = e - N_EDGES; }
  for (int hd = 0; hd < heads; ++hd) {
    float v = al_s[s * heads + hd] + al_d[d * heads + hd];
    v = v > 0.f ? v : NEG_SLOPE * v;
    alpha[(size_t)e * heads + hd] = v;
    atomicMax(&mx[d * heads + hd], v);   // global_atomic_max_num_f32
  }
}

// e = exp(alpha - m[dst]) (stored in-place over alpha); segment sum via atomicAdd
__global__ void edge_expsum(const long long* __restrict__ edst, const float* __restrict__ mx,
                            float* __restrict__ alpha, float* __restrict__ ssum, int heads) {
  int e = blockIdx.x * blockDim.x + threadIdx.x;
  if (e >= E_TOT) return;
  int d = (e < N_EDGES) ? (int)edst[e] : e - N_EDGES;
  for (int hd = 0; hd < heads; ++hd) {
    size_t ai = (size_t)e * heads + hd;
    float ev = expf(alpha[ai] - mx[d * heads + hd]);
    alpha[ai] = ev;
    atomicAdd(&ssum[d * heads + hd], ev);
  }
}

// out[dst,h,:] += h[src,h,:] * e/(s[dst]+1e-16); one thread per (edge, head)
__global__ void edge_aggregate(const long long* __restrict__ esrc,
                               const long long* __restrict__ edst,
                               const float* __restrict__ ev, const float* __restrict__ ssum,
                               const float* __restrict__ h, float* __restrict__ out,
                               int heads, int C) {
  int idx = blockIdx.x * blockDim.x + threadIdx.x;
  if (idx >= E_TOT * heads) return;
  int e = idx / heads, hd = idx - e * heads;
  int s, d;
  if (e < N_EDGES) { s = (int)esrc[e]; d = (int)edst[e]; }
  else             { s = d = e - N_EDGES; }
  float w = ev[idx] / (ssum[d * heads + hd] + 1e-16f);
  const float* hp = h + ((size_t)s * heads + hd) * C;
  float* op = out + ((size_t)d * heads + hd) * C;
  for (int c = 0; c < C; ++c) atomicAdd(&op[c], hp[c] * w);
}

__global__ void bias_act(float* __restrict__ out, const float* __restrict__ b,
                         int n, int F, int relu) {
  int i = blockIdx.x * blockDim.x + threadIdx.x;
  if (i >= n * F) return;
  float v = out[i] + b[i % F];
  if (relu) v = v > 0.f ? v : 0.f;
  out[i] = v;
}

// global_max_pool over batch ids (batch sorted, int64), 4 channels
__global__ void pool_max(const float* __restrict__ h, const long long* __restrict__ batch,
                         float* __restrict__ g) {
  int i = blockIdx.x * blockDim.x + threadIdx.x;
  if (i >= N_NODES * 4) return;
  int node = i >> 2, c = i & 3;
  int gi = (int)batch[node];
  atomicMax(&g[gi * 4 + c], h[i]);
}

__global__ void log_softmax4(const float* __restrict__ g, float* __restrict__ out) {
  int r = blockIdx.x * blockDim.x + threadIdx.x;
  if (r >= N_GRAPHS) return;
  float m = g[r * 4];
  for (int c = 1; c < 4; ++c) m = fmaxf(m, g[r * 4 + c]);
  float s = 0.f;
  for (int c = 0; c < 4; ++c) s += expf(g[r * 4 + c] - m);
  float ls = logf(s);
  for (int c = 0; c < 4; ++c) out[r * 4 + c] = g[r * 4 + c] - m - ls;
}

// ---------------------------------------------------------------------------
template <int F, int H, int C>
static void run_layer(const float* x, const long long* esrc, const long long* edst,
                      const float* W, const float* asv, const float* adv, const float* b,
                      int relu,
                      float* hbuf, float* outb, float* alS, float* alD,
                      float* mbuf, float* sbuf, float* ebuf, hipStream_t stream) {
  constexpr int NH = H * C;
  const int n = N_NODES;
  const int tiles = (n / 16) * ((NH + 15) / 16);
  wmma_gemm_f32<F, NH><<<(tiles + 7) / 8, 256, 0, stream>>>(x, W, hbuf, n);
  node_logits<<<(n * H + 255) / 256, 256, 0, stream>>>(hbuf, asv, adv, alS, alD, n, H, C);
  // NOTE: outb may alias x; stream order guarantees the GEMM (only reader of x)
  // has retired before this fill executes.
  fill_f32<<<(n * H + 255) / 256, 256, 0, stream>>>(mbuf, -INFINITY, n * H);
  fill_f32<<<(n * H + 255) / 256, 256, 0, stream>>>(sbuf, 0.f, n * H);
  fill_f32<<<(n * NH + 255) / 256, 256, 0, stream>>>(outb, 0.f, n * NH);
  edge_alpha_max<<<(E_TOT + 255) / 256, 256, 0, stream>>>(esrc, edst, alS, alD, ebuf, mbuf, H);
  edge_expsum<<<(E_TOT + 255) / 256, 256, 0, stream>>>(edst, mbuf, ebuf, sbuf, H);
  edge_aggregate<<<(E_TOT * H + 255) / 256, 256, 0, stream>>>(esrc, edst, ebuf, sbuf,
                                                              hbuf, outb, H, C);
  bias_act<<<(n * NH + 255) / 256, 256, 0, stream>>>(outb, b, n, NH, relu);
}

extern "C" void kernel_launch(void* const* d_in, const int* in_sizes, int n_in,
                              void* d_out, int out_size, void* d_ws, size_t ws_size,
                              hipStream_t stream) {
  (void)in_sizes; (void)n_in; (void)out_size; (void)ws_size;
  const float*     x     = (const float*)d_in[0];
  const long long* eidx  = (const long long*)d_in[1];
  const long long* batch = (const long long*)d_in[2];
  const float* W1 = (const float*)d_in[3];
  const float* aS1 = (const float*)d_in[4];
  const float* aD1 = (const float*)d_in[5];
  const float* b1 = (const float*)d_in[6];
  const float* W2 = (const float*)d_in[7];
  const float* aS2 = (const float*)d_in[8];
  const float* aD2 = (const float*)d_in[9];
  const float* b2 = (const float*)d_in[10];
  const float* W3 = (const float*)d_in[11];
  const float* aS3 = (const float*)d_in[12];
  const float* aD3 = (const float*)d_in[13];
  const float* b3 = (const float*)d_in[14];
  const long long* esrc = eidx;
  const long long* edst = eidx + N_EDGES;

  // Workspace carve (all f32, ~107 MB total)
  size_t off = 0;
  char* ws = (char*)d_ws;
  auto carve = [&](size_t elems) { float* p = (float*)(ws + off); off += elems * sizeof(float); return p; };
  float* hbuf = carve((size_t)N_NODES * 135);       // post-GEMM features
  float* xbuf = carve((size_t)N_NODES * 135);       // aggregated out / next-layer x
  float* alS  = carve((size_t)N_NODES * 3);
  float* alD  = carve((size_t)N_NODES * 3);
  float* mbuf = carve((size_t)N_NODES * 3);
  float* sbuf = carve((size_t)N_NODES * 3);
  float* ebuf = carve((size_t)E_TOT * 3);
  float* gbuf = carve((size_t)N_GRAPHS * 4);

  // Layer 1: 9 -> 3 heads x 45, ReLU
  run_layer<9, 3, 45>(x,    esrc, edst, W1, aS1, aD1, b1, 1,
                      hbuf, xbuf, alS, alD, mbuf, sbuf, ebuf, stream);
  // Layer 2: 135 -> 3 heads x 18, ReLU   (xbuf aliases input; safe, see note)
  run_layer<135, 3, 18>(xbuf, esrc, edst, W2, aS2, aD2, b2, 1,
                        hbuf, xbuf, alS, alD, mbuf, sbuf, ebuf, stream);
  // Layer 3: 54 -> 1 head x 4, no ReLU
  run_layer<54, 1, 4>(xbuf, esrc, edst, W3, aS3, aD3, b3, 0,
                      hbuf, xbuf, alS, alD, mbuf, sbuf, ebuf, stream);

  // Graph max-pool (init -1e9 == neg-inf guard) + log_softmax
  fill_f32<<<(N_GRAPHS * 4 + 255) / 256, 256, 0, stream>>>(gbuf, -1e9f, N_GRAPHS * 4);
  pool_max<<<(N_NODES * 4 + 255) / 256, 256, 0, stream>>>(xbuf, batch, gbuf);
  log_softmax4<<<(N_GRAPHS + 255) / 256, 256, 0, stream>>>(gbuf, (float*)d_out);
}